// Tsukamoto_13769665151063
// MI455X (gfx1250) — compile-verified
//
#include <hip/hip_runtime.h>
#include <cstdint>

#define B_  8
#define S_  4096
#define R_  16
#define M_  5
#define NX_ 200

typedef __attribute__((ext_vector_type(2))) float        v2f;
typedef __attribute__((ext_vector_type(8))) float        v8f;
typedef __attribute__((ext_vector_type(4))) unsigned int v4u;
typedef __attribute__((ext_vector_type(8))) int          v8i;
typedef __attribute__((ext_vector_type(4))) int          v4i;

// ---------------------------------------------------------------------------
// Kernel 1: build function_rules[R, NX] = max over selected memberships of
// exp(-0.5*((x-c)/s)^2). 3200 elements, one-shot, transcendental-heavy.
// ---------------------------------------------------------------------------
__global__ void build_rules_kernel(const float* __restrict__ centers,
                                   const float* __restrict__ sigmas,
                                   const int*   __restrict__ rules,
                                   float*       __restrict__ fr_out) {
  int i = blockIdx.x * blockDim.x + threadIdx.x;
  if (i >= R_ * NX_) return;
  int r  = i / NX_;
  int nx = i - r * NX_;
  float x = (float)nx * (1.0f / (float)(NX_ - 1));
  float best = 0.0f;
#pragma unroll
  for (int m = 0; m < M_; ++m) {
    float z = (x - centers[m]) / sigmas[m];
    float v = __expf(-0.5f * z * z);
    best = (rules[r * M_ + m] == 1) ? fmaxf(best, v) : best;
  }
  fr_out[i] = best;
}

// ---------------------------------------------------------------------------
// Kernel 2: per (b,s): acc[16x16] += A(min(fr, f)) [16 rules x 4 nx]
//                                  x B([X | 1 | 0...]) [4 nx x 16]
// via V_WMMA_F32_16X16X4_F32, 50 K-steps covering NX=200.
// D column 0 (lanes 0/16) = Sum Y*X per rule, column 1 (lanes 1/17) = Sum Y.
// fr table staged into LDS with the Tensor Data Mover (TENSORcnt path).
// ---------------------------------------------------------------------------
__global__ __launch_bounds__(256) void
tsukamoto_wmma_kernel(const float* __restrict__ f,
                      const float* __restrict__ fr_g,
                      float*       __restrict__ out,
                      int nbs) {
  __shared__ float fr_s[R_ * NX_];

#if defined(__gfx1250__) && __has_builtin(__builtin_amdgcn_tensor_load_to_lds)
  if (threadIdx.x < 32) {  // wave 0 issues the TDM op (EXEC-all-1 within wave)
    unsigned           lds_addr = (unsigned)(unsigned long long)(void*)&fr_s[0];
    unsigned long long ga       = (unsigned long long)(const void*)fr_g;
    v4u g0;
    g0[0] = 1u;                                               // count=1 (valid user D#)
    g0[1] = lds_addr;                                         // lds_addr [63:32]
    g0[2] = (unsigned)(ga & 0xFFFFFFFFu);                     // global_addr [95:64]
    g0[3] = (unsigned)((ga >> 32) & 0x1FFFFFFu) | 0x80000000u;// addr[56:32] | type=2
    v8i g1;
    g1[0] = 0x00020000;            // data_size=2 -> 4 bytes; no multicast/barrier
    g1[1] = (int)(200u << 16);     // tensor_dim0 = 200  (bits 79:48, low half)
    g1[2] = (int)(16u  << 16);     // tensor_dim1 = 16   (bits 111:80, low half)
    g1[3] = (int)(200u << 16);     // tile_dim0   = 200  (bits 127:112)
    g1[4] = 16;                    // tile_dim1   = 16   (bits 143:128)
    g1[5] = 200;                   // tensor_dim0_stride = 200 (bits 207:160)
    g1[6] = 0;
    g1[7] = 0;
    v4i g2 = {0, 0, 0, 0};         // unused (2-D tensor)
    v4i g3 = {0, 0, 0, 0};
    v8i g4 = {0, 0, 0, 0, 0, 0, 0, 0};  // clang-23 extended group (unused)
    __builtin_amdgcn_tensor_load_to_lds(g0, g1, g2, g3, g4, 0);
#if __has_builtin(__builtin_amdgcn_s_wait_tensorcnt)
    __builtin_amdgcn_s_wait_tensorcnt(0);
#endif
  }
#else
  for (int i = threadIdx.x; i < R_ * NX_; i += 256) fr_s[i] = fr_g[i];
#endif
  __syncthreads();

  const int   lane = threadIdx.x & 31;
  const int   wv   = threadIdx.x >> 5;
  const int   row  = lane & 15;   // A: rule (M); B: column (N)
  const int   half = lane >> 4;   // A/B: K sub-pair selector
  const float dx   = 1.0f / (float)(NX_ - 1);
  // B column selectors: N==0 -> X weights, N==1 -> ones, else 0
  const float xsel = (row == 0) ? 1.0f : 0.0f;
  const float csel = (row == 1) ? 1.0f : 0.0f;

  for (int bs = blockIdx.x * 8 + wv; bs < nbs; bs += gridDim.x * 8) {
    const float  fval = f[bs * R_ + row];              // firing strength, rule=row
    const float* frp  = &fr_s[row * NX_ + 2 * half];   // rule row, K-pair base
    v8f acc = {};
    float bx = (float)(2 * half) * dx;                 // X at this lane's first K

#if defined(__gfx1250__) && __has_builtin(__builtin_amdgcn_wmma_f32_16x16x4_f32)
#pragma unroll 5
    for (int nx0 = 0; nx0 < NX_; nx0 += 4) {
      v2f A, Bv;
      A.x  = fminf(frp[nx0    ], fval);                // Y = min(consequent, f)
      A.y  = fminf(frp[nx0 + 1], fval);
      Bv.x = xsel * (bx)      + csel;                  // [X | 1 | 0...] columns
      Bv.y = xsel * (bx + dx) + csel;
      acc  = __builtin_amdgcn_wmma_f32_16x16x4_f32(
                 false, A, false, Bv, (short)0, acc, false, false);
      bx += 4.0f * dx;
    }
    float s = acc[0] + acc[1] + acc[2] + acc[3] +
              acc[4] + acc[5] + acc[6] + acc[7];       // fold M within lane
    float num = __shfl(s, 0, 32) + __shfl(s, 16, 32);  // column N=0
    float den = __shfl(s, 1, 32) + __shfl(s, 17, 32);  // column N=1
#else
    // VALU fallback (kept only so a missing builtin can't fail the build)
    float num = 0.0f, den = 0.0f;
    for (int nx = 2 * half; nx < NX_; nx += 4) {
      float y0 = fminf(frp[nx - 2 * half + 0], fval);  // unreachable on gfx1250
      float y1 = fminf(frp[nx - 2 * half + 1], fval);
      num += y0 * (nx * dx) + y1 * ((nx + 1) * dx);
      den += y0 + y1;
    }
    for (int o = 16; o > 0; o >>= 1) { num += __shfl_down(num, o, 32); den += __shfl_down(den, o, 32); }
#endif
    if (lane == 0) out[bs] = num / den;
  }
}

// ---------------------------------------------------------------------------
extern "C" void kernel_launch(void* const* d_in, const int* in_sizes, int n_in,
                              void* d_out, int out_size, void* d_ws, size_t ws_size,
                              hipStream_t stream) {
  const float* f       = (const float*)d_in[0];  // [B,S,R]
  const float* centers = (const float*)d_in[1];  // [M]
  const float* sigmas  = (const float*)d_in[2];  // [M]
  const int*   rules   = (const int*)  d_in[3];  // [R,M]
  float*       out     = (float*)d_out;          // [B,S,1]
  float*       fr      = (float*)d_ws;           // [R,NX] scratch (12.8 KB)

  (void)in_sizes; (void)n_in; (void)ws_size;

  const int tab = R_ * NX_;
  build_rules_kernel<<<(tab + 255) / 256, 256, 0, stream>>>(centers, sigmas, rules, fr);

  const int nbs = B_ * S_;  // 32768 (b,s) pairs; one per wave-tile, grid-stride
  tsukamoto_wmma_kernel<<<512, 256, 0, stream>>>(f, fr, out, nbs);
}